// GHMC_38680475467827
// MI455X (gfx1250) — compile-verified
//
#include <hip/hip_runtime.h>

#define BINS 10
#define NCNT (BINS + 1)   // 10 bin counters + total-valid counter
#define TPB  256
#define MAXBLOCKS 2048ULL

// Exact vector type the async builtin expects for both pointer args
typedef int v4i __attribute__((vector_size(4 * sizeof(int))));

// CDNA5 async global->LDS staging path (guarded: falls back to plain b128 loads)
#if defined(__AMDGCN__) && __has_builtin(__builtin_amdgcn_global_load_async_to_lds_b128) && __has_builtin(__builtin_amdgcn_s_wait_asynccnt)
#define GHM_ASYNC 1
#define GHM_GPTR(p) ((__attribute__((address_space(1))) v4i*)(p))
#define GHM_LPTR(p) ((__attribute__((address_space(3))) v4i*)(p))
#else
#define GHM_ASYNC 0
#endif

__device__ __forceinline__ void ghm_bin(float p, float w, bool& valid, bool& inb, int& idx) {
  valid = (w > 0.0f);
  float g = fabsf(expf(-p) - 1.0f);          // |exp(-pred) - 1|
  int b = (int)floorf(g * (float)BINS);      // floor(g*bins)
  b = (b < 0) ? 0 : b;
  b = (b > BINS - 1) ? (BINS - 1) : b;       // clip to [0, bins-1]
  idx = b;
  inb = valid && (g < (1.0f + 1e-6f));       // g < edges[-1]
}

// Wave-uniform counting: ballot result is an SGPR, so the accumulators stay
// scalar. Only lane 0 of each wave later publishes them. Lane 0 always has the
// smallest index in a wave, so it executes every iteration the wave executes.
__device__ __forceinline__ void ghm_count(float p, float w, unsigned* c) {
  bool valid, inb; int idx;
  ghm_bin(p, w, valid, inb, idx);
#if defined(__AMDGCN__)
  c[BINS] += (unsigned)__builtin_popcount(__builtin_amdgcn_ballot_w32(valid));
#pragma unroll
  for (int k = 0; k < BINS; ++k)
    c[k] += (unsigned)__builtin_popcount(__builtin_amdgcn_ballot_w32(inb && (idx == k)));
#endif
}

// ---------------- Pass 1: global histogram ----------------
__global__ void __launch_bounds__(TPB)
ghm_hist_kernel(const float4* __restrict__ pred4, const float4* __restrict__ lw4,
                unsigned long long n4, unsigned long long ntot,
                unsigned* __restrict__ gcounts) {
  __shared__ unsigned sh[NCNT];
  if (threadIdx.x < NCNT) sh[threadIdx.x] = 0u;
  __syncthreads();

  unsigned c[NCNT];
#pragma unroll
  for (int k = 0; k < NCNT; ++k) c[k] = 0u;

  const unsigned long long stride = (unsigned long long)gridDim.x * TPB;
  for (unsigned long long i = (unsigned long long)blockIdx.x * TPB + threadIdx.x;
       i < n4; i += stride) {
    if (i + stride < n4) {
      __builtin_prefetch(&pred4[i + stride], 0, 3);   // global_prefetch_b8
      __builtin_prefetch(&lw4[i + stride], 0, 3);
    }
    float4 p = pred4[i];
    float4 w = lw4[i];
    float pe[4] = {p.x, p.y, p.z, p.w};
    float we[4] = {w.x, w.y, w.z, w.w};
#pragma unroll
    for (int e = 0; e < 4; ++e) ghm_count(pe[e], we[e], c);
  }

  // scalar tail (ntot not multiple of 4)
  {
    const float* ps = (const float*)pred4;
    const float* ws = (const float*)lw4;
    for (unsigned long long t = 4ULL * n4 + (unsigned long long)blockIdx.x * TPB + threadIdx.x;
         t < ntot; t += stride)
      ghm_count(ps[t], ws[t], c);
  }

  if ((threadIdx.x & 31u) == 0u) {
#pragma unroll
    for (int k = 0; k < NCNT; ++k)
      if (c[k]) atomicAdd(&sh[k], c[k]);        // ds_add_u32
  }
  __syncthreads();
  if (threadIdx.x < NCNT) {
    unsigned v = sh[threadIdx.x];
    if (v) atomicAdd(&gcounts[threadIdx.x], v); // global_atomic_add_u32
  }
}

// ---------------- Pass 2: apply weights ----------------
__device__ __forceinline__ float ghm_weight(float p, float w, const float* wbin) {
  bool valid, inb; int idx;
  ghm_bin(p, w, valid, inb, idx);
  float wt = wbin[idx];                  // LDS table lookup
  return inb ? wt : 0.0f;
}

__global__ void __launch_bounds__(TPB)
ghm_apply_kernel(const float4* __restrict__ pred4, const float4* __restrict__ lw4,
                 const unsigned* __restrict__ gcounts, float4* __restrict__ out4,
                 unsigned long long n4, unsigned long long ntot) {
  __shared__ float wbin[BINS];
#if GHM_ASYNC
  __shared__ float4 sp[2][TPB];
  __shared__ float4 sw[2][TPB];
#endif

  if (threadIdx.x == 0) {
    float tot = fmaxf((float)gcounts[BINS], 1.0f);
    unsigned cnt[BINS];
    int n = 0;
#pragma unroll
    for (int k = 0; k < BINS; ++k) { cnt[k] = gcounts[k]; n += (cnt[k] > 0u) ? 1 : 0; }
    float nf = fmaxf((float)n, 1.0f);
#pragma unroll
    for (int k = 0; k < BINS; ++k)
      // matches reference rounding: (tot / counts) then (/ n), two f32 divides
      wbin[k] = (cnt[k] > 0u) ? ((tot / (float)cnt[k]) / nf) : 0.0f;
  }
  __syncthreads();

  const unsigned long long stride = (unsigned long long)gridDim.x * TPB;
  const unsigned long long i0 = (unsigned long long)blockIdx.x * TPB + threadIdx.x;

#if GHM_ASYNC
  // Double-buffered async global->LDS staging. Each lane stages only its own
  // 16B slot, so no cross-wave barrier is needed — just s_wait_asynccnt.
  int buf = 0;
  if (i0 < n4) {
    __builtin_amdgcn_global_load_async_to_lds_b128(
        GHM_GPTR(pred4 + i0), GHM_LPTR(&sp[0][threadIdx.x]), 0, 0);
    __builtin_amdgcn_global_load_async_to_lds_b128(
        GHM_GPTR(lw4 + i0), GHM_LPTR(&sw[0][threadIdx.x]), 0, 0);
  }
  for (unsigned long long i = i0; i < n4; i += stride) {
    unsigned long long inext = i + stride;
    if (inext < n4) {
      __builtin_amdgcn_global_load_async_to_lds_b128(
          GHM_GPTR(pred4 + inext), GHM_LPTR(&sp[buf ^ 1][threadIdx.x]), 0, 0);
      __builtin_amdgcn_global_load_async_to_lds_b128(
          GHM_GPTR(lw4 + inext), GHM_LPTR(&sw[buf ^ 1][threadIdx.x]), 0, 0);
      __builtin_amdgcn_s_wait_asynccnt(2);   // current pair landed (in-order)
    } else {
      __builtin_amdgcn_s_wait_asynccnt(0);
    }
    float4 p = sp[buf][threadIdx.x];
    float4 w = sw[buf][threadIdx.x];
    float4 r;
    r.x = ghm_weight(p.x, w.x, wbin) * p.x;
    r.y = ghm_weight(p.y, w.y, wbin) * p.y;
    r.z = ghm_weight(p.z, w.z, wbin) * p.z;
    r.w = ghm_weight(p.w, w.w, wbin) * p.w;
    out4[i] = r;
    buf ^= 1;
  }
#else
  for (unsigned long long i = i0; i < n4; i += stride) {
    float4 p = pred4[i];
    float4 w = lw4[i];
    float4 r;
    r.x = ghm_weight(p.x, w.x, wbin) * p.x;
    r.y = ghm_weight(p.y, w.y, wbin) * p.y;
    r.z = ghm_weight(p.z, w.z, wbin) * p.z;
    r.w = ghm_weight(p.w, w.w, wbin) * p.w;
    out4[i] = r;
  }
#endif

  // scalar tail
  {
    const float* ps = (const float*)pred4;
    const float* ws = (const float*)lw4;
    float* os = (float*)out4;
    for (unsigned long long t = 4ULL * n4 + i0; t < ntot; t += stride)
      os[t] = ghm_weight(ps[t], ws[t], wbin) * ps[t];
  }
}

extern "C" void kernel_launch(void* const* d_in, const int* in_sizes, int n_in,
                              void* d_out, int out_size, void* d_ws, size_t ws_size,
                              hipStream_t stream) {
  (void)n_in; (void)out_size; (void)ws_size;
  const float* pred = (const float*)d_in[0];
  // d_in[1] = target (unused by the math)
  const float* lw   = (const float*)d_in[2];
  float* out = (float*)d_out;

  const unsigned long long ntot = (unsigned long long)in_sizes[0];
  const unsigned long long n4   = ntot / 4ULL;
  unsigned* gcounts = (unsigned*)d_ws;

  // zero the 11 counters every launch (graph-capture-safe async memset)
  (void)hipMemsetAsync(d_ws, 0, NCNT * sizeof(unsigned), stream);

  unsigned long long want = (n4 + TPB - 1) / TPB;
  if (want < 1) want = 1;
  int blocks = (int)((want < MAXBLOCKS) ? want : MAXBLOCKS);

  ghm_hist_kernel<<<dim3(blocks), dim3(TPB), 0, stream>>>(
      (const float4*)pred, (const float4*)lw, n4, ntot, gcounts);
  ghm_apply_kernel<<<dim3(blocks), dim3(TPB), 0, stream>>>(
      (const float4*)pred, (const float4*)lw, gcounts, (float4*)out, n4, ntot);
}